// FastTritonAdaptivePiecewiseConv2d_88519275970726
// MI455X (gfx1250) — compile-verified
//
#include <hip/hip_runtime.h>

// ---------------------------------------------------------------------------
// FastTritonAdaptivePiecewiseConv2d for MI455X (gfx1250, wave32)
//   x:         [4, 32, 32, 32]  f32
//   positions: [32, 32, 3, 3, 5] f32   (unsorted breakpoints)
//   values:    [32, 32, 3, 3, 5] f32
//   out:       [4, 32, 32, 32]  f32
//
// Per-(o,tap) piecewise-linear LUT => no matrix-op factorization (WMMA not
// applicable). CDNA5 path used: Tensor Data Mover (tensor_load_to_lds) to
// stage the per-channel tables into LDS + s_wait_tensorcnt, ds_load_b128
// broadcast reads in the hot loop.
// ---------------------------------------------------------------------------

#define B_   4
#define C_   32
#define O_   32
#define HW_  32
#define P_   5
#define T_   288          // C_ * 3 * 3
#define TP_  1440         // T_ * P_
#define EPS_ 1e-6f

typedef unsigned int u32x4 __attribute__((ext_vector_type(4)));
typedef int          i32x8 __attribute__((ext_vector_type(8)));
typedef int          i32x4 __attribute__((ext_vector_type(4)));

#if defined(__has_builtin)
#  if __has_builtin(__builtin_amdgcn_tensor_load_to_lds)
#    define HAVE_TDM 1
#  endif
#endif
#ifndef HAVE_TDM
#  define HAVE_TDM 0
#endif

// Low 32 bits of a generic pointer into LDS == byte offset within the
// workgroup's LDS allocation (aperture layout, ISA 10.2).
__device__ __forceinline__ unsigned lds_off(const void* p) {
    return (unsigned)(size_t)p;
}

#if HAVE_TDM
// Build a 1-D Tensor DMA descriptor (D#, ISA 8.3-8.6) copying `nelem` dwords
// from `gsrc` to LDS byte offset `lds_byte`, and issue TENSOR_LOAD_TO_LDS.
// clang-23 / therock-10.0 arity: (v4u, v8i, v4i, v4i, v8i, i32 cpol).
__device__ __forceinline__ void tdm_copy_1d(const float* gsrc, unsigned lds_byte,
                                            unsigned nelem) {
    unsigned long long ga = (unsigned long long)(size_t)gsrc;
    u32x4 g0;
    g0[0] = 1u;                                   // count=1, user mode, no gather
    g0[1] = lds_byte;                             // lds_addr [63:32]
    g0[2] = (unsigned)(ga & 0xFFFFFFFFu);         // global_addr[31:0]
    g0[3] = (unsigned)((ga >> 32) & 0x1FFFFFFu)   // global_addr[56:32]
          | (2u << 30);                           // type=2 ("image")
    i32x8 g1;
    g1[0] = (int)(2u << 16);                      // data_size=2 (4B); wg_mask=0
    g1[1] = (int)((nelem & 0xFFFFu) << 16);       // tensor_dim0[15:0] @ [63:48]
    g1[2] = (int)(((nelem >> 16) & 0xFFFFu)       // tensor_dim0[31:16]
          | (1u << 16));                          // tensor_dim1 = 1
    g1[3] = (int)((nelem & 0xFFFFu) << 16);       // tile_dim0 @ [127:112]
    g1[4] = 1;                                    // tile_dim1 = 1, tile_dim2 = 0
    g1[5] = (int)nelem;                           // tensor_dim0_stride[31:0]
    g1[6] = (int)((nelem & 0xFFFFu) << 16);       // dim0_stride hi=0; dim1_stride lo
    g1[7] = 0;                                    // tensor_dim1_stride hi
    i32x4 g2;
    g2[0] = 1;                                    // tensor_dim2
    g2[1] = 1;                                    // tensor_dim3
    g2[2] = 0;                                    // tensor_dim2_stride lo
    g2[3] = 0;                                    // stride hi, tile_dim3 = 0
    i32x4 g3 = {0, 0, 0, 0};                      // dim3_stride/dim4/tile_dim4 = 0
    i32x8 g4 = {0, 0, 0, 0, 0, 0, 0, 0};          // extra group (zero-filled)
    __builtin_amdgcn_tensor_load_to_lds(g0, g1, g2, g3, g4, 0);
}
#endif

__global__ __launch_bounds__(256) void
pwconv_gfx1250_kernel(const float* __restrict__ x,
                      const float* __restrict__ positions,
                      const float* __restrict__ values,
                      float* __restrict__ out) {
    // LDS: raw staging (TDM target), then sorted stride-8 tables + reciprocals
    __shared__ __align__(16) float sh_raw[2 * TP_];   // 11520 B
    __shared__ __align__(16) float sh_pos8[T_ * 8];   //  9216 B
    __shared__ __align__(16) float sh_val8[T_ * 8];   //  9216 B
    __shared__ __align__(16) float sh_inv4[T_ * 4];   //  4608 B

    const int tid  = threadIdx.x;
    const int bid  = blockIdx.x;
    const int tile = bid & 3;          // 8-row strip
    const int o    = (bid >> 2) & 31;  // output channel
    const int b    = bid >> 7;         // batch

    const float* pos_g = positions + (size_t)o * TP_;
    const float* val_g = values    + (size_t)o * TP_;

#if HAVE_TDM
    if (tid < 32) {                    // wave 0 drives the Tensor Data Mover
        tdm_copy_1d(pos_g, lds_off(&sh_raw[0]),   TP_);
        tdm_copy_1d(val_g, lds_off(&sh_raw[TP_]), TP_);
        __builtin_amdgcn_s_wait_tensorcnt(0);
    }
#else
    for (int k = tid; k < TP_; k += 256) {
        sh_raw[k]       = pos_g[k];
        sh_raw[TP_ + k] = val_g[k];
    }
#endif
    __syncthreads();

    // Sort each tap's 5 (pos,val) pairs by pos (Bose-Nelson 9-CE network ==
    // stable result for distinct keys) and precompute 1/(dp+eps) per segment.
    for (int tap = tid; tap < T_; tap += 256) {
        float p[5], v[5];
#pragma unroll
        for (int k = 0; k < 5; ++k) {
            p[k] = sh_raw[tap * 5 + k];
            v[k] = sh_raw[TP_ + tap * 5 + k];
        }
#define CE(a, c)                                                            \
        { if (p[a] > p[c]) { float tp = p[a]; p[a] = p[c]; p[c] = tp;       \
                             float tv = v[a]; v[a] = v[c]; v[c] = tv; } }
        CE(0, 1) CE(3, 4) CE(2, 4) CE(2, 3) CE(0, 3)
        CE(0, 2) CE(1, 4) CE(1, 3) CE(1, 2)
#undef CE
#pragma unroll
        for (int k = 0; k < 5; ++k) {
            sh_pos8[tap * 8 + k] = p[k];
            sh_val8[tap * 8 + k] = v[k];
        }
#pragma unroll
        for (int k = 0; k < 4; ++k)
            sh_inv4[tap * 4 + k] = 1.0f / (p[k + 1] - p[k] + EPS_);
    }
    __syncthreads();

    const int w = tid & 31;                 // lane -> column (coalesced)
    const int h = tile * 8 + (tid >> 5);    // wave -> row (uniform per wave)
    const float* xb = x + (size_t)b * (C_ * HW_ * HW_);

    float acc = 0.0f;
#pragma unroll
    for (int i = 0; i < 3; ++i) {
        const int hy = h + i - 1;
#pragma unroll
        for (int j = 0; j < 3; ++j) {
            const int wx  = w + j - 1;
            const bool inb = ((unsigned)hy < 32u) && ((unsigned)wx < 32u);
            const float* xp = xb + hy * HW_ + wx;   // stride C: 1024 floats
            const int tbase = i * 3 + j;
#pragma unroll 4
            for (int c = 0; c < C_; ++c) {
                const float xv  = inb ? xp[c * 1024] : 0.0f;  // zero padding
                const int   tap = c * 9 + tbase;              // wave-uniform
                // broadcast ds_load_b128 x3 + two b32
                const float4 pv = *(const float4*)(sh_pos8 + tap * 8);
                const float  p4 = sh_pos8[tap * 8 + 4];
                const float4 vv = *(const float4*)(sh_val8 + tap * 8);
                const float  v4 = sh_val8[tap * 8 + 4];
                const float4 iv = *(const float4*)(sh_inv4 + tap * 4);

                // Sorted breakpoints => segment select is monotone in the
                // compare masks: idx = (x>=p1)+(x>=p2)+(x>=p3). Gather each
                // quantity with a 3-deep cndmask chain reusing b1..b3.
                const bool b1 = (xv >= pv.y);
                const bool b2 = (xv >= pv.z);
                const bool b3 = (xv >= pv.w);
                const float pj  = b3 ? pv.w : b2 ? pv.z : b1 ? pv.y : pv.x;
                const float pj1 = b3 ? p4   : b2 ? pv.w : b1 ? pv.z : pv.y;
                const float vj  = b3 ? vv.w : b2 ? vv.z : b1 ? vv.y : vv.x;
                const float vj1 = b3 ? v4   : b2 ? vv.w : b1 ? vv.z : vv.y;
                const float ivj = b3 ? iv.w : b2 ? iv.z : b1 ? iv.y : iv.x;

                float t = (xv - pj) * ivj;                 // no per-tap divide
                t = t < 0.0f ? 0.0f : (t > 1.0f ? 1.0f : t);
                float r = fmaf(t, vj1 - vj, vj);
                r = (pj == pj1) ? vj   : r;   // degenerate segment
                r = (xv >= p4)  ? v4   : r;   // clamp high
                r = (xv <= pv.x)? vv.x : r;   // clamp low (highest precedence)
                acc += r;
            }
        }
    }
    out[(((size_t)b * O_ + o) * HW_ + h) * HW_ + w] = acc;
}

extern "C" void kernel_launch(void* const* d_in, const int* in_sizes, int n_in,
                              void* d_out, int out_size, void* d_ws, size_t ws_size,
                              hipStream_t stream) {
    (void)in_sizes; (void)n_in; (void)out_size; (void)d_ws; (void)ws_size;
    const float* x   = (const float*)d_in[0];
    const float* pos = (const float*)d_in[1];
    const float* val = (const float*)d_in[2];
    float* out = (float*)d_out;
    dim3 grid(B_ * O_ * 4);   // 512 blocks: (b, o, 8-row strip)
    dim3 block(256);          // 8 wave32s
    hipLaunchKernelGGL(pwconv_gfx1250_kernel, grid, block, 0, stream,
                       x, pos, val, out);
}